// ChunkOperator_62783831933059
// MI455X (gfx1250) — compile-verified
//
#include <hip/hip_runtime.h>

// out[b, m, c, f] = x[b, m*125 + c, f]
//   B=16, T=32000, F=64, CHUNK=250, ADV=125, M=255
// Pure overlapping-window gather: bandwidth-bound (≈392 MB HBM traffic,
// ~17 us roofline at 23.3 TB/s). No FLOPs -> WMMA inapplicable; the optimal
// CDNA5 path is wave32 b128 coalesced loads + non-temporal b128 stores
// (output is write-once; NT keeps it from evicting the reused input in L2).

typedef __attribute__((ext_vector_type(4))) float v4f;

constexpr int B   = 16;
constexpr int T   = 32000;   // SIG_LEN
constexpr int F   = 64;      // N_FILT
constexpr int C   = 250;     // CHUNK_LEN
constexpr int ADV = 125;     // CHUNK_ADV
constexpr int M   = 255;     // NUM_OVERLAP
constexpr int V   = F / 4;   // 16 float4 per frame row
constexpr int NV  = C * V;   // 4000 float4 per output chunk (64 KB)

__global__ __launch_bounds__(256) void chunk_overlap_kernel(
    const v4f* __restrict__ in, v4f* __restrict__ out) {
  const int m = blockIdx.x;   // output frame index  [0, 255)
  const int b = blockIdx.y;   // batch               [0, 16)

  // Each frame is a contiguous 250*64-float span of the input starting at
  // sample m*ADV; the whole block is a shifted 64 KB contiguous copy.
  const v4f* __restrict__ src = in  + ((size_t)b * T + (size_t)m * ADV) * V;
  v4f*       __restrict__ dst = out + ((size_t)b * M + (size_t)m) * (size_t)NV;

  #pragma unroll 4
  for (int i = (int)threadIdx.x; i < NV; i += 256) {
    v4f val = src[i];                        // global_load_b128 (RT: reused by m+1)
    __builtin_nontemporal_store(val, &dst[i]); // global_store_b128 TH=NT (write-once)
  }
}

extern "C" void kernel_launch(void* const* d_in, const int* in_sizes, int n_in,
                              void* d_out, int out_size, void* d_ws, size_t ws_size,
                              hipStream_t stream) {
  (void)in_sizes; (void)n_in; (void)d_ws; (void)ws_size; (void)out_size;
  const v4f* in  = (const v4f*)d_in[0];
  v4f*       out = (v4f*)d_out;

  dim3 grid(M, B);          // 255 x 16 = 4080 blocks, 64 KB copy each
  dim3 block(256);          // 8 wave32 waves per block
  chunk_overlap_kernel<<<grid, block, 0, stream>>>(in, out);
}